// SGCModel_88948772700672
// MI455X (gfx1250) — compile-verified
//
#include <hip/hip_runtime.h>

#define N_NODES   100000
#define EMBED_DIM 128
#define N_EDGES   3200000

typedef __attribute__((ext_vector_type(2))) float v2f;
typedef __attribute__((ext_vector_type(8))) float v8f;

// ---------------------------------------------------------------- utilities

__global__ void fill_zero_kernel(float* __restrict__ p, int n) {
    int i = blockIdx.x * blockDim.x + threadIdx.x;
    if (i < n) p[i] = 0.0f;
}

// deg[dst[e]] += 1.0  (3.2M edges, E multiple of 256 -> no bounds check)
__global__ void degree_kernel(const int* __restrict__ dst, float* __restrict__ deg) {
    int e = blockIdx.x * blockDim.x + threadIdx.x;
    atomicAdd(&deg[dst[e]], 1.0f);
}

// in-place: deg -> rsqrt(max(deg,1))
__global__ void norm_kernel(float* __restrict__ dn, int n) {
    int i = blockIdx.x * blockDim.x + threadIdx.x;
    if (i < n) dn[i] = rsqrtf(fmaxf(dn[i], 1.0f));
}

// out[node,:] = in[node,:] * norm[node]; one thread per float4 (32 groups/node)
__global__ void scale_kernel(const float* __restrict__ in,
                             const float* __restrict__ nrm,
                             float* __restrict__ out) {
    int t    = blockIdx.x * blockDim.x + threadIdx.x;   // [0, N_NODES*32)
    int node = t >> 5;
    int grp  = t & 31;
    float s  = nrm[node];
    float4 v = ((const float4*)(in + node * EMBED_DIM))[grp];
    v.x *= s; v.y *= s; v.z *= s; v.w *= s;
    ((float4*)(out + node * EMBED_DIM))[grp] = v;
}

// hdst[dst[e],:] += hsrc[src[e],:]; one thread per (edge, float4-group)
__global__ void scatter_kernel(const float* __restrict__ hsrc,
                               const int* __restrict__ src,
                               const int* __restrict__ dst,
                               float* __restrict__ hdst) {
    int t   = blockIdx.x * blockDim.x + threadIdx.x;    // [0, N_EDGES*32)
    int e   = t >> 5;
    int grp = t & 31;
    int s = src[e];
    int d = dst[e];
    float4 v  = ((const float4*)(hsrc + s * EMBED_DIM))[grp];
    float* dp = hdst + d * EMBED_DIM + grp * 4;
    atomicAdd(dp + 0, v.x);
    atomicAdd(dp + 1, v.y);
    atomicAdd(dp + 2, v.z);
    atomicAdd(dp + 3, v.w);
}

// ------------------------------------------------- linear layer via f32 WMMA
// Hout[i][j] = sum_k Hin[i][k] * W[j][k] + b[j]
// A = Hin 16x4 tile (M x K), B = W^T 4x16 tile (K x N), D = 16x16 f32.
// Layouts per CDNA5 ISA 7.12.2:
//   A (32-bit, 16x4): lanes 0-15 hold M=lane, VGPR0=K0/VGPR1=K1;
//                     lanes 16-31 hold M=lane-16, K2/K3.
//   B mirrored: N striped across lanes, K across VGPRs/lane-halves.
//   C/D: VGPR v, lane l -> M = v + (l>=16 ? 8 : 0), N = l&15.
__global__ void __launch_bounds__(256)
sgc_linear_wmma_kernel(const float* __restrict__ Hin,
                       const float* __restrict__ Wm,
                       const float* __restrict__ bias,
                       float* __restrict__ Hout) {
    const int lane  = threadIdx.x & 31;
    const int wave  = threadIdx.x >> 5;        // 0..7 -> N tile
    const int mbase = blockIdx.x * 16;         // 6250 exact tiles
    const int nbase = wave * 16;
    const int r     = lane & 15;               // A: row M; B: col N
    const int khalf = (lane >> 4) << 1;        // 0 (lanes 0-15) or 2 (lanes 16-31)

    const float* arow = Hin + (mbase + r) * EMBED_DIM;   // A[M=r][k]
    const float* brow = Wm  + (nbase + r) * EMBED_DIM;   // B[k][N=r] = W[N=r][k]

    v8f c = {};
#pragma unroll
    for (int k = 0; k < EMBED_DIM; k += 4) {
        v2f a, b;
        a.x = arow[k + khalf];
        a.y = arow[k + khalf + 1];
        b.x = brow[k + khalf];
        b.y = brow[k + khalf + 1];
        // 8 args: (neg_a, A, neg_b, B, c_mod, C, reuse_a, reuse_b)
        c = __builtin_amdgcn_wmma_f32_16x16x4_f32(
                false, a, false, b, (short)0, c, false, false);
    }

    float bv = bias[nbase + r];
#pragma unroll
    for (int v = 0; v < 8; ++v) {
        int M = v + ((lane >> 4) << 3);
        Hout[(mbase + M) * EMBED_DIM + nbase + r] = c[v] + bv;
    }
}

// ------------------------------------------------------------ edge dot (u·v)
// One wave per edge: 32 lanes x float4 cover 128 features; shfl_xor reduce.
__global__ void edge_dot_kernel(const float* __restrict__ h,
                                const int* __restrict__ s_idx,
                                const int* __restrict__ d_idx,
                                float* __restrict__ out) {
    int t    = blockIdx.x * blockDim.x + threadIdx.x;   // [0, N_EDGES*32)
    int e    = t >> 5;
    int lane = t & 31;
    int s = s_idx[e];
    int d = d_idx[e];
    float4 a = ((const float4*)(h + s * EMBED_DIM))[lane];
    float4 b = ((const float4*)(h + d * EMBED_DIM))[lane];
    float p = a.x * b.x + a.y * b.y + a.z * b.z + a.w * b.w;
#pragma unroll
    for (int off = 16; off > 0; off >>= 1)
        p += __shfl_xor(p, off, 32);
    if (lane == 0) out[e] = p;
}

// --------------------------------------------------------------------- host

extern "C" void kernel_launch(void* const* d_in, const int* in_sizes, int n_in,
                              void* d_out, int out_size, void* d_ws, size_t ws_size,
                              hipStream_t stream) {
    const float* emb  = (const float*)d_in[0];   // [N, 128]
    const float* Wm   = (const float*)d_in[1];   // [128, 128]
    const float* bias = (const float*)d_in[2];   // [128]
    const int*   src  = (const int*)d_in[3];     // [E]
    const int*   dst  = (const int*)d_in[4];     // [E]
    const int*   nsrc = (const int*)d_in[5];     // [E]
    const int*   ndst = (const int*)d_in[6];     // [E]
    float* out = (float*)d_out;                  // [2E]

    const size_t NF = (size_t)N_NODES * EMBED_DIM;   // 12.8M floats
    float* hA  = (float*)d_ws;                       // current h
    float* hB  = hA + NF;                            // scaled / final h
    float* nrm = hB + NF;                            // deg -> norm  [N]

    const int TB = 256;
    const int blkN    = (N_NODES + TB - 1) / TB;         // 391
    const int blkNF   = (int)(NF / TB);                  // 50000
    const int blkScal = N_NODES * 32 / TB;               // 12500
    const int blkE    = N_EDGES / TB;                    // 12500
    const int blkEF   = (int)((size_t)N_EDGES * 32 / TB); // 400000

    // degree -> norm
    fill_zero_kernel<<<blkN, TB, 0, stream>>>(nrm, N_NODES);
    degree_kernel<<<blkE, TB, 0, stream>>>(dst, nrm);
    norm_kernel<<<blkN, TB, 0, stream>>>(nrm, N_NODES);

    // hop 1: hB = emb*norm ; hA = scatter(hB) ; hA *= norm
    scale_kernel<<<blkScal, TB, 0, stream>>>(emb, nrm, hB);
    fill_zero_kernel<<<blkNF, TB, 0, stream>>>(hA, (int)NF);
    scatter_kernel<<<blkEF, TB, 0, stream>>>(hB, src, dst, hA);
    scale_kernel<<<blkScal, TB, 0, stream>>>(hA, nrm, hA);

    // hop 2
    scale_kernel<<<blkScal, TB, 0, stream>>>(hA, nrm, hB);
    fill_zero_kernel<<<blkNF, TB, 0, stream>>>(hA, (int)NF);
    scatter_kernel<<<blkEF, TB, 0, stream>>>(hB, src, dst, hA);
    scale_kernel<<<blkScal, TB, 0, stream>>>(hA, nrm, hA);

    // linear: hB = hA @ W^T + b  (WMMA, one 16x16 tile per wave)
    sgc_linear_wmma_kernel<<<N_NODES / 16, 256, 0, stream>>>(hA, Wm, bias, hB);

    // edge dots: positive then negative
    edge_dot_kernel<<<blkEF, TB, 0, stream>>>(hB, src, dst, out);
    edge_dot_kernel<<<blkEF, TB, 0, stream>>>(hB, nsrc, ndst, out + N_EDGES);
}